// ROIPooling_2121713844583
// MI455X (gfx1250) — compile-verified
//
#include <hip/hip_runtime.h>
#include <cstdint>
#include <cstddef>

// Problem constants (from reference): B,C,L,N,S
#define B_ 32
#define C_ 256
#define L_ 512
#define N_ 1024
#define S_ 8

#define CH_PER_BLOCK 64   // channels handled per workgroup (grid.y = C_/CH_PER_BLOCK)
#define CH_CHUNK     16   // channels staged per LDS buffer (<= 16*512*4B = 32 KB)
#define CH_PER_WAVE  2    // rows moved by each wave's TDM descriptor (8 waves * 2 = 16)
#define NCHUNK       (CH_PER_BLOCK / CH_CHUNK)   // 4 chunks, double-buffered

typedef unsigned int u32x4 __attribute__((ext_vector_type(4)));
typedef unsigned int u32x8 __attribute__((ext_vector_type(8)));

__global__ __launch_bounds__(256) void roi_pool_tdm_db_kernel(
    const float* __restrict__ sentences,   // (B, C, L)
    const int*   __restrict__ rois,        // (N, 2)
    const int*   __restrict__ roi_idx,     // (N,)
    float*       __restrict__ out)         // (N, C, S)
{
    __shared__ float tile[2 * CH_CHUNK * L_];   // 2 x 32 KB double buffer

    const int n  = blockIdx.x;             // ROI index
    const int cq = blockIdx.y;             // channel quarter
    const int t  = threadIdx.x;

    const int b  = roi_idx[n];
    const int x1 = rois[2 * n + 0];
    const int x2 = rois[2 * n + 1];
    const int lr = x2 - x1;

    // 16B-aligned staging range [p_lo, p_lo+cnt) covering [x1, x2)
    const int p_lo = x1 & ~3;
    int p_hi = (x2 + 3) & ~3;
    if (p_hi > L_) p_hi = L_;
    const int cnt = p_hi - p_lo;           // multiple of 4, 4..512

    // ---- compute-phase mapping: (channel, bin, half) ----
    const int ch_l = t >> 4;               // 0..15 channel within chunk
    const int bin  = (t >> 1) & 7;         // 0..7
    const int half = t & 1;                // 0..1

    const int bstart = x1 + (bin * lr) / S_;
    const int bend   = x1 + ((bin + 1) * lr + S_ - 1) / S_;

    // ---- TDM mapping: each wave DMAs CH_PER_WAVE rows of a chunk ----
    const int wave_id = t >> 5;                        // 0..7 (wave32)
    const int w_ch0   = wave_id * CH_PER_WAVE;         // first row this wave moves
    const unsigned lds_base =
        (unsigned)(uintptr_t)(&tile[0]);               // LDS byte offset of tile

    const int c_base = cq * CH_PER_BLOCK;

    // Issue one TDM descriptor for chunk `ci` into buffer ci&1.
    // Tile: tile_dim0 = cnt elems (row), tile_dim1 = 2 rows, row stride = L_.
    // LDS rows written back-to-back (compacted): row r -> buf[(w_ch0+r)*cnt ...]
    auto issue_tdm = [&](int ci) {
        const int c0 = c_base + ci * CH_CHUNK;
        unsigned long long gaddr = (unsigned long long)(uintptr_t)
            (sentences + ((size_t)b * C_ + (size_t)(c0 + w_ch0)) * L_ + (size_t)p_lo);
        unsigned lds_addr = lds_base
                          + (unsigned)(ci & 1) * (CH_CHUNK * L_ * 4u)
                          + (unsigned)(w_ch0 * cnt) * 4u;

        // D# group 0 (128b): count=1 | lds_addr | global_addr[56:0] | type=2
        u32x4 g0;
        g0.x = 1u;                                          // count = 1 valid D#
        g0.y = lds_addr;                                    // LDS byte address
        g0.z = (unsigned)(gaddr & 0xFFFFFFFFull);           // global_addr[31:0]
        g0.w = (unsigned)((gaddr >> 32) & 0x01FFFFFFull)    // global_addr[56:32]
               | (2u << 30);                                // type = 2 ("image")

        // D# group 1 (256b)
        u32x8 g1;
        g1.s0 = (2u << 16);                                 // data_size=2 (4B); mask=0
        g1.s1 = ((unsigned)cnt & 0xFFFFu) << 16;            // tensor_dim0[15:0]
        g1.s2 = ((unsigned)cnt >> 16)                       // tensor_dim0[31:16]
                | ((unsigned)CH_PER_WAVE << 16);            // tensor_dim1[15:0]
        g1.s3 = ((unsigned)cnt << 16);                      // tile_dim0 = cnt
        g1.s4 = (unsigned)CH_PER_WAVE;                      // tile_dim1=2, tile_dim2=0
        g1.s5 = (unsigned)L_;                               // tensor_dim0_stride lo
        g1.s6 = 0u;                                         // stride0 hi | stride1 lo
        g1.s7 = 0u;                                         // stride1 hi

        // 2-group form: VADDR2/VADDR3 = NULL -> up-to-2D tensor
        asm volatile("tensor_load_to_lds %0, %1"
                     :: "s"(g0), "s"(g1)
                     : "memory");
    };

    // -------- software pipeline: TDM for chunk i+1 overlaps compute of chunk i ----
    issue_tdm(0);

    for (int ci = 0; ci < NCHUNK; ++ci) {
        if (ci + 1 < NCHUNK) {
            issue_tdm(ci + 1);                      // prefetch next chunk (other buffer)
            __builtin_amdgcn_s_wait_tensorcnt(1);   // per-wave in-order: chunk ci done
        } else {
            __builtin_amdgcn_s_wait_tensorcnt(0);   // last chunk: drain
        }
        __syncthreads();                            // all waves' chunk-ci tiles visible

        // ---- segmented max over this thread's (channel, bin, half) ----
        const float* tb = &tile[(ci & 1) * (CH_CHUNK * L_)];
        float m = -INFINITY;
        const int base = ch_l * cnt - p_lo;         // compacted row layout
        for (int p = bstart + half; p < bend; p += 2) {
            m = fmaxf(m, tb[base + p]);
        }
        m = fmaxf(m, __shfl_xor(m, 1, 32));         // combine halves (wave32)
        if (half == 0) {
            const int c0 = c_base + ci * CH_CHUNK;
            out[((size_t)n * C_ + (size_t)(c0 + ch_l)) * S_ + bin] = m;
        }
        __syncthreads();   // buf ci&1 free before TDM(ci+2) overwrites it
    }
}

extern "C" void kernel_launch(void* const* d_in, const int* in_sizes, int n_in,
                              void* d_out, int out_size, void* d_ws, size_t ws_size,
                              hipStream_t stream) {
    const float* sentences = (const float*)d_in[0];  // (B,C,L,1) float32
    const int*   rois      = (const int*)  d_in[1];  // (N,2) integer
    const int*   roi_idx   = (const int*)  d_in[2];  // (N,)  integer
    float*       out       = (float*)d_out;          // (N,C,S,1) float32

    dim3 grid(N_, C_ / CH_PER_BLOCK);   // 1024 x 4 workgroups
    dim3 block(256);                    // 8 wave32 waves
    roi_pool_tdm_db_kernel<<<grid, block, 0, stream>>>(sentences, rois, roi_idx, out);
}